// PositionLoss_56556129354082
// MI455X (gfx1250) — compile-verified
//
#include <hip/hip_runtime.h>
#include <math.h>

typedef __attribute__((ext_vector_type(2))) float v2f;
typedef __attribute__((ext_vector_type(8))) float v8f;

#define PTS_PER_BATCH 4096
#define CHUNK 2048
#define NTILES 128   // CHUNK / 16

__global__ void chamfer_zero(float* accum) {
    if (threadIdx.x == 0) accum[0] = 0.0f;
}

// One wave = 16 rows of A-side points vs ALL 2048 B-side points of one
// (dir, chunk, batch).
//
// dist^2[m][n] = |x_m|^2 + (-2 x.y + |y_n|^2); the parenthesized part is one
// V_WMMA_F32_16X16X4_F32 per 16x16 tile with the spare K=3 slot carrying the
// (1.0, |y_n|^2) pair, so C is the inline-0 literal and no broadcast movs are
// needed. B operands are pre-staged in LDS in native B-matrix layout:
//   lanes  0-15 (K0,K1) read plane 0: (-2*y0, -2*y1)
//   lanes 16-31 (K2,K3) read plane 1: (-2*y2, |y|^2)
__global__ __launch_bounds__(256) void chamfer_wmma(const float* __restrict__ x,
                                                    const float* __restrict__ y,
                                                    float* __restrict__ accum) {
    __shared__ float2 sB[2 * CHUNK];   // 32 KB: two planes of float2 per point

    const int g     = blockIdx.x;        // 0..511
    const int tg    = g & 15;            // group of 8 row-tiles
    const int b     = (g >> 4) & 7;      // batch
    const int chunk = (g >> 7) & 1;
    const int dir   = (g >> 8) & 1;      // 0: x->y min, 1: y->x min
    const int p0    = chunk * CHUNK;

    const float* Apts = dir ? y : x;
    const float* Bpts = dir ? x : y;

    // Cooperative stage of all B-side points for this (chunk, batch)
    for (int k = 0; k < CHUNK / 256; ++k) {
        int p   = threadIdx.x + k * 256;
        int idx = (b * PTS_PER_BATCH + p0 + p) * 3;
        float y0 = Bpts[idx + 0];
        float y1 = Bpts[idx + 1];
        float y2 = Bpts[idx + 2];
        float yy = y0 * y0 + y1 * y1 + y2 * y2;
        sB[p]         = make_float2(-2.0f * y0, -2.0f * y1);  // K0, K1
        sB[p + CHUNK] = make_float2(-2.0f * y2, yy);          // K2, K3
    }
    __syncthreads();

    const int wave = threadIdx.x >> 5;
    const int lane = threadIdx.x & 31;
    const int half = lane >> 4;          // 0: lanes 0-15, 1: lanes 16-31
    const int lm   = lane & 15;
    const int rowTile = tg * 8 + wave;

    // A matrix 16x4 f32: lane lm holds row M=lm; lanes 0-15: (K0,K1),
    // lanes 16-31: (K2,K3).  K3 = 1.0 pairs with B's K3 = |y|^2.
    int m    = p0 + rowTile * 16 + lm;
    int aidx = (b * PTS_PER_BATCH + m) * 3;
    float a0 = Apts[aidx + 0];
    float a1 = Apts[aidx + 1];
    float a2 = Apts[aidx + 2];
    float xx = a0 * a0 + a1 * a1 + a2 * a2;

    v2f avec;
    avec[0] = half ? a2 : a0;
    avec[1] = half ? 1.0f : a1;

    // Per-lane base into the staged B planes (computed once)
    const float2* laneB = &sB[half * CHUNK + lm];

    v8f rowmin;
#pragma unroll
    for (int i = 0; i < 8; ++i) rowmin[i] = 3.0e38f;

    for (int t = 0; t < NTILES; ++t) {
        float2 q = laneB[t * 16];        // one ds_load_b64 per lane
        v2f bvec;
        bvec[0] = q.x;
        bvec[1] = q.y;
        v8f c = {};                      // lowers to inline literal 0

        // D[m][n] = -2*dot(x_m, y_n) + |y_n|^2
        v8f d = __builtin_amdgcn_wmma_f32_16x16x4_f32(
            false, avec, false, bvec, (short)0, c, false, false);

#pragma unroll
        for (int i = 0; i < 8; ++i) rowmin[i] = fminf(rowmin[i], d[i]);
    }

    // C/D layout: VGPR i, lanes 0-15 -> M=i, lanes 16-31 -> M=i+8, N=lane%16.
    // Min over N = butterfly across the 16 lanes sharing a row.
#pragma unroll
    for (int off = 1; off < 16; off <<= 1) {
#pragma unroll
        for (int i = 0; i < 8; ++i)
            rowmin[i] = fminf(rowmin[i], __shfl_xor(rowmin[i], off, 32));
    }

    // Add |x_m|^2 back, sqrt once per row, sum the 16 rows of this strip.
    float partial = 0.0f;
#pragma unroll
    for (int i = 0; i < 8; ++i) {
        float xxv = __shfl(xx, i + 8 * half, 32);  // xx of row (i + 8*half)
        float d2  = xxv + rowmin[i];
        partial += sqrtf(fmaxf(d2, 0.0f));
    }
    partial += __shfl_xor(partial, 16, 32);        // combine both half-groups

    if (lane == 0) atomicAdd(accum, partial);
}

__global__ void chamfer_final(const float* __restrict__ accum,
                              float* __restrict__ out) {
    // Every min-distance row carries weight 1/(8*2048) in the loss.
    if (threadIdx.x == 0) out[0] = accum[0] * (1.0f / 16384.0f);
}

extern "C" void kernel_launch(void* const* d_in, const int* in_sizes, int n_in,
                              void* d_out, int out_size, void* d_ws, size_t ws_size,
                              hipStream_t stream) {
    (void)in_sizes; (void)n_in; (void)out_size; (void)ws_size;
    const float* x = (const float*)d_in[0];
    const float* y = (const float*)d_in[1];
    float* out   = (float*)d_out;
    float* accum = (float*)d_ws;

    chamfer_zero<<<1, 32, 0, stream>>>(accum);
    // 4096 strips = 2 dirs * 2 chunks * 8 batches * 128 row-tiles; 8 waves/block
    chamfer_wmma<<<512, 256, 0, stream>>>(x, y, accum);
    chamfer_final<<<1, 32, 0, stream>>>(accum, out);
}